// PoolPointsInterp_38843684225258
// MI455X (gfx1250) — compile-verified
//
#include <hip/hip_runtime.h>
#include <cstdint>
#include <cstddef>

// Problem constants (fixed by the reference's setup_inputs).
constexpr int   Nb = 8;
constexpr int   Cc = 256;
constexpr int   Hh = 168;
constexpr int   Ww = 168;
constexpr int   Ss = Hh * Ww;            // 28224 spatial positions
constexpr float kScale = 0.125f;

typedef float v4f __attribute__((ext_vector_type(4)));

// ---------------------------------------------------------------------------
// Per-point bilinear setup, matching the JAX reference exactly.
// ---------------------------------------------------------------------------
struct PointW {
  int b, xl, xh, yl, yh;
  float w1, w2, w3, w4;
};

__device__ __forceinline__ PointW point_weights(const float* __restrict__ rois, int p) {
  const float rb = rois[3 * p + 0];
  const float rx = rois[3 * p + 1];
  const float ry = rois[3 * p + 2];
  float x = rx * kScale;
  float y = ry * kScale;
  const bool valid = !((y < -1.0f) || (y > (float)Hh) || (x < -1.0f) || (x > (float)Ww));
  x = fmaxf(x, 0.0f);
  y = fmaxf(y, 0.0f);
  int xl = (int)x;
  int yl = (int)y;
  int xh, yh;
  if (xl >= Ww - 1) { xl = Ww - 1; xh = Ww - 1; x = (float)xl; } else { xh = xl + 1; }
  if (yl >= Hh - 1) { yl = Hh - 1; yh = Hh - 1; y = (float)yl; } else { yh = yl + 1; }
  const float lx = x - (float)xl;
  const float ly = y - (float)yl;
  const float hx = 1.0f - lx;
  const float hy = 1.0f - ly;
  const float vm = valid ? 1.0f : 0.0f;
  PointW w;
  w.b = (int)rb;
  w.xl = xl; w.xh = xh; w.yl = yl; w.yh = yh;
  w.w1 = hy * hx * vm;
  w.w2 = hy * lx * vm;
  w.w3 = ly * hx * vm;
  w.w4 = ly * lx * vm;
  return w;
}

// ---------------------------------------------------------------------------
// Kernel 1: NCHW -> NHWC transpose of the whole feature tensor into d_ws.
// Per (b, c-tile, s-tile) block: 32x32 tile. Phase 1 streams the tile into
// LDS with gfx1250 async global->LDS loads (ASYNCcnt-tracked, per-lane LDS
// destination, non-temporal: feat is read exactly once, so don't let it
// evict the freshly written NHWC lines from the 192MB L2); phase 2 reads
// LDS transposed (stride-33, bank-conflict-free) and writes coalesced NHWC
// rows (regular temporal: re-read by the gather pass).
// ---------------------------------------------------------------------------
__global__ __launch_bounds__(256) void ppi_transpose_nchw_to_nhwc(
    const float* __restrict__ feat, float* __restrict__ out) {
  __shared__ float tile[32][33];
  const int st   = blockIdx.x;  // 0..881  (s tiles of 32)
  const int ct   = blockIdx.y;  // 0..7    (c tiles of 32)
  const int b    = blockIdx.z;  // 0..7
  const int t    = threadIdx.x;
  const int lane = t & 31;
  const int row8 = t >> 5;      // 0..7

  const float* src = feat + ((size_t)(b * Cc + ct * 32) * (size_t)Ss + (size_t)st * 32);

#pragma unroll
  for (int i = 0; i < 4; ++i) {
    const int r = row8 + i * 8;                       // channel row within tile
    const float* gp = src + (size_t)r * Ss + lane;    // 128B-contiguous per wave
    const unsigned ldsa = (unsigned)(uintptr_t)(&tile[r][lane]);
    // gfx1250 async copy of 4 bytes from the global address in vaddr to the LDS address in vdst; tracked by ASYNCcnt.
    asm volatile("global_load_async_to_lds_b32 %0, %1, off th:TH_LOAD_NT"
                 :: "v"(ldsa), "v"(gp) : "memory");
  }
  asm volatile("s_wait_asynccnt 0" ::: "memory");
  __syncthreads();

  float* dst = out + ((size_t)b * Ss + (size_t)st * 32) * (size_t)Cc + (size_t)ct * 32;
#pragma unroll
  for (int i = 0; i < 4; ++i) {
    const int sr = row8 + i * 8;                      // spatial row within tile
    dst[(size_t)sr * Cc + lane] = tile[lane][sr];     // coalesced 128B store
  }
}

// ---------------------------------------------------------------------------
// Kernel 2: gather from NHWC. One wave per point; lane l covers channel
// float4s {l, l+32}: 8x b128 loads (4 corners x 2), FMA blend, 2x b128 NT
// stores. Everything is fully coalesced 512B-per-wave-instruction traffic.
// ---------------------------------------------------------------------------
__global__ __launch_bounds__(256) void ppi_gather_nhwc(
    const float* __restrict__ nhwc, const float* __restrict__ rois,
    float* __restrict__ out, int P) {
  const int wave = threadIdx.x >> 5;
  const int lane = threadIdx.x & 31;
  const int p = blockIdx.x * 8 + wave;
  if (p >= P) return;

  const PointW w = point_weights(rois, p);

  const float* base = nhwc + (size_t)w.b * (size_t)Ss * Cc;
  const v4f* c1 = (const v4f*)(base + ((size_t)w.yl * Ww + w.xl) * Cc);
  const v4f* c2 = (const v4f*)(base + ((size_t)w.yl * Ww + w.xh) * Cc);
  const v4f* c3 = (const v4f*)(base + ((size_t)w.yh * Ww + w.xl) * Cc);
  const v4f* c4 = (const v4f*)(base + ((size_t)w.yh * Ww + w.xh) * Cc);
  v4f* o = (v4f*)(out + (size_t)p * Cc);

#pragma unroll
  for (int k = 0; k < 2; ++k) {
    const int ci = lane + k * 32;   // float4 index: covers 256 channels
    const v4f a = c1[ci];
    const v4f bb = c2[ci];
    const v4f cc = c3[ci];
    const v4f dd = c4[ci];
    v4f r = w.w1 * a + w.w2 * bb + w.w3 * cc + w.w4 * dd;
    __builtin_nontemporal_store(r, &o[ci]);   // output is never re-read
  }
}

// ---------------------------------------------------------------------------
// Fallback: direct NCHW gather (used only if d_ws is too small for the
// transposed tensor). Wave per point; lane strides channels.
// ---------------------------------------------------------------------------
__global__ __launch_bounds__(256) void ppi_gather_nchw(
    const float* __restrict__ feat, const float* __restrict__ rois,
    float* __restrict__ out, int P) {
  const int wave = threadIdx.x >> 5;
  const int lane = threadIdx.x & 31;
  const int p = blockIdx.x * 8 + wave;
  if (p >= P) return;

  const PointW w = point_weights(rois, p);
  const float* fb = feat + (size_t)w.b * (size_t)Cc * Ss;
  const int o11 = w.yl * Ww + w.xl;
  const int o12 = w.yl * Ww + w.xh;
  const int o21 = w.yh * Ww + w.xl;
  const int o22 = w.yh * Ww + w.xh;

#pragma unroll
  for (int k = 0; k < 8; ++k) {
    const int c = lane + k * 32;
    const float* fc = fb + (size_t)c * Ss;
    const float r = w.w1 * fc[o11] + w.w2 * fc[o12] + w.w3 * fc[o21] + w.w4 * fc[o22];
    __builtin_nontemporal_store(r, &out[(size_t)p * Cc + c]);
  }
}

// ---------------------------------------------------------------------------
extern "C" void kernel_launch(void* const* d_in, const int* in_sizes, int n_in,
                              void* d_out, int out_size, void* d_ws, size_t ws_size,
                              hipStream_t stream) {
  const float* feat = (const float*)d_in[0];
  const float* rois = (const float*)d_in[1];
  float* out = (float*)d_out;
  const int P = in_sizes[1] / 3;
  const int gather_blocks = (P + 7) / 8;

  const size_t nhwc_bytes = (size_t)Nb * Cc * Ss * sizeof(float);  // ~231 MB

  if (ws_size >= nhwc_bytes) {
    float* nhwc = (float*)d_ws;
    dim3 tgrid(Ss / 32, Cc / 32, Nb);  // (882, 8, 8)
    ppi_transpose_nchw_to_nhwc<<<tgrid, 256, 0, stream>>>(feat, nhwc);
    ppi_gather_nhwc<<<gather_blocks, 256, 0, stream>>>(nhwc, rois, out, P);
  } else {
    ppi_gather_nchw<<<gather_blocks, 256, 0, stream>>>(feat, rois, out, P);
  }
}